// RPEMultiHeadAttention_395136991669
// MI455X (gfx1250) — compile-verified
//
#include <hip/hip_runtime.h>
#include <math.h>

// ---------------- problem constants ----------------
#define K_DMODEL 512
#define K_NHEAD  8
#define K_HDIM   64
#define K_B      4
#define K_T      2048
#define K_S      2048
#define K_MAXREL 128

typedef float v2f __attribute__((ext_vector_type(2)));
typedef float v8f __attribute__((ext_vector_type(8)));

static __device__ __forceinline__ v8f wmma_f32(v2f a, v2f b, v8f c) {
  // V_WMMA_F32_16X16X4_F32 : D(16x16 f32) = A(16x4 f32) * B(4x16 f32) + C
  return __builtin_amdgcn_wmma_f32_16x16x4_f32(false, a, false, b, (short)0, c,
                                               false, false);
}

// =====================================================================
// Kernel 1: fused Q/K/V projection.  Y = X @ W^T + b -> [B, H, L, Dh].
// One wave computes a 32x32 output tile (2x2 sub-tiles of 16x16) so each
// loaded A/B fragment feeds two WMMAs.
// Fragment layouts (f32 16x16x4):
//   A (16x4 MxK):  lane(half,l15): M=l15, VGPRs = K{2*half, 2*half+1}
//   B (4x16 KxN):  lane(half,l15): N=l15, VGPRs = K{2*half, 2*half+1}
//   (B[k][n] = W[n0+n][k0+k] -> contiguous float2 of the W row)
// =====================================================================
__global__ __launch_bounds__(128) void proj_qkv_kernel(
    const float* __restrict__ query, const float* __restrict__ key,
    const float* __restrict__ value,
    const float* __restrict__ Wq, const float* __restrict__ bq,
    const float* __restrict__ Wk, const float* __restrict__ bk,
    const float* __restrict__ Wv, const float* __restrict__ bv,
    float* __restrict__ qb, float* __restrict__ kb, float* __restrict__ vb) {
  const int wave = threadIdx.x >> 5;
  const int lane = threadIdx.x & 31;
  const int half = lane >> 4;
  const int l15  = lane & 15;

  const int m0 = (blockIdx.x * 4 + wave) * 32;  // 0..8160
  const int n0 = blockIdx.y * 32;               // 0..480
  const int which = blockIdx.z;                 // 0:q 1:k 2:v

  const float* X; const float* W; const float* bias; float* Y;
  if (which == 0)      { X = query; W = Wq; bias = bq; Y = qb; }
  else if (which == 1) { X = key;   W = Wk; bias = bk; Y = kb; }
  else                 { X = value; W = Wv; bias = bv; Y = vb; }

  const float* xr0 = X + (size_t)(m0 + l15) * K_DMODEL + 2 * half;
  const float* xr1 = xr0 + (size_t)16 * K_DMODEL;
  const float* wr0 = W + (size_t)(n0 + l15) * K_DMODEL + 2 * half;
  const float* wr1 = wr0 + (size_t)16 * K_DMODEL;

  v8f acc00 = {}, acc01 = {}, acc10 = {}, acc11 = {};
#pragma unroll 4
  for (int k0 = 0; k0 < K_DMODEL; k0 += 4) {
    v2f a0 = *(const v2f*)(xr0 + k0);
    v2f a1 = *(const v2f*)(xr1 + k0);
    v2f b0 = *(const v2f*)(wr0 + k0);
    v2f b1 = *(const v2f*)(wr1 + k0);
    acc00 = wmma_f32(a0, b0, acc00);
    acc01 = wmma_f32(a0, b1, acc01);
    acc10 = wmma_f32(a1, b0, acc10);
    acc11 = wmma_f32(a1, b1, acc11);
  }

#pragma unroll
  for (int mi = 0; mi < 2; ++mi) {
#pragma unroll
    for (int ni = 0; ni < 2; ++ni) {
      const v8f acc = (mi == 0) ? (ni == 0 ? acc00 : acc01)
                                : (ni == 0 ? acc10 : acc11);
      const int col = n0 + ni * 16 + l15;
      const int h   = col >> 6;
      const int dh  = col & 63;
      const float bn = bias[col];
#pragma unroll
      for (int r = 0; r < 8; ++r) {
        const int row = m0 + mi * 16 + r + 8 * half;  // C: M = r + 8*half
        const int bb  = row >> 11;
        const int pos = row & 2047;
        Y[(((size_t)(bb * K_NHEAD + h)) * K_T + pos) * K_HDIM + dh] =
            acc[r] + bn;
      }
    }
  }
}

// =====================================================================
// Kernel 2: flash attention (online softmax).  One wave owns TWO 16-row
// query tiles (32 rows) of one (b,h): K/V fragments loaded once per
// 16-key step feed both tiles' WMMAs.  rel_emb staged in LDS; tiles
// fully outside the +-128 clamp window take a uniform-bias fast path.
// =====================================================================
__global__ __launch_bounds__(128) void flash_attn_kernel(
    const float* __restrict__ qb, const float* __restrict__ kb,
    const float* __restrict__ vb, const float* __restrict__ rel_emb,
    float* __restrict__ ctx) {
  __shared__ float srel[(2 * K_MAXREL + 1) * K_NHEAD];  // 8224 B
  __shared__ float pbuf[4][2][16][17];                  // P staging per wave

  for (int i = threadIdx.x; i < (2 * K_MAXREL + 1) * K_NHEAD; i += 128)
    srel[i] = rel_emb[i];
  __syncthreads();

  const int wave = threadIdx.x >> 5;
  const int lane = threadIdx.x & 31;
  const int half = lane >> 4;
  const int l15  = lane & 15;

  const int pair  = blockIdx.x * 4 + wave;  // 0..2047
  const int bh    = pair >> 6;              // b*NHEAD + h
  const int tpair = pair & 63;
  const int h     = bh & (K_NHEAD - 1);
  const int b     = bh >> 3;
  const int t0    = tpair * 32;             // first of 32 query rows

  const float* Q = qb + ((size_t)bh * K_T + t0) * K_HDIM;
  const float* K = kb + (size_t)bh * K_S * K_HDIM;
  const float* V = vb + (size_t)bh * K_S * K_HDIM;

  // Q A-fragments for both 16-row sub-tiles
  v2f qa0[16], qa1[16];
  {
    const float* qr0 = Q + l15 * K_HDIM + 2 * half;
    const float* qr1 = qr0 + 16 * K_HDIM;
#pragma unroll
    for (int kk = 0; kk < 16; ++kk) {
      qa0[kk] = *(const v2f*)(qr0 + 4 * kk);
      qa1[kk] = *(const v2f*)(qr1 + 4 * kk);
    }
  }

  float mrow[2][8], lrow[2][8];
  v8f o0[4] = {}, o1[4] = {};
#pragma unroll
  for (int r = 0; r < 8; ++r) {
    mrow[0][r] = -INFINITY; lrow[0][r] = 0.0f;
    mrow[1][r] = -INFINITY; lrow[1][r] = 0.0f;
  }

  const float scale    = 0.125f;  // 1/sqrt(64)
  const float bias_lo  = srel[0 * K_NHEAD + h];            // d <= -128
  const float bias_hi  = srel[2 * K_MAXREL * K_NHEAD + h]; // d >= +128

  for (int s0 = 0; s0 < K_S; s0 += 16) {
    // ---- scores: both tiles share the K fragments ----
    v8f sc0 = {}, sc1 = {};
    const float* krow = K + (size_t)(s0 + l15) * K_HDIM + 2 * half;
#pragma unroll
    for (int kk = 0; kk < 16; ++kk) {
      v2f kf = *(const v2f*)(krow + 4 * kk);
      sc0 = wmma_f32(qa0[kk], kf, sc0);
      sc1 = wmma_f32(qa1[kk], kf, sc1);
    }

    // ---- bias (uniform fast path when whole 32x16 tile clamps) ----
    const int scol = s0 + l15;
    const bool uni_hi = (t0 - (s0 + 15)) >= K_MAXREL;       // all d >= +128
    const bool uni_lo = (s0 - (t0 + 31)) >= K_MAXREL;       // all d <= -128
    float p[2][8];
#pragma unroll
    for (int sub = 0; sub < 2; ++sub) {
      const v8f sc = sub ? sc1 : sc0;
      if (uni_hi || uni_lo) {
        const float bc = uni_hi ? bias_hi : bias_lo;
#pragma unroll
        for (int r = 0; r < 8; ++r) p[sub][r] = sc[r] * scale + bc;
      } else {
#pragma unroll
        for (int r = 0; r < 8; ++r) {
          const int t = t0 + sub * 16 + r + 8 * half;
          int d = t - scol;
          d = (d < -K_MAXREL) ? -K_MAXREL : (d > K_MAXREL ? K_MAXREL : d);
          p[sub][r] = sc[r] * scale + srel[(d + K_MAXREL) * K_NHEAD + h];
        }
      }
    }

    // ---- online softmax per sub-tile ----
#pragma unroll
    for (int sub = 0; sub < 2; ++sub) {
      float rmax[8];
#pragma unroll
      for (int r = 0; r < 8; ++r) rmax[r] = p[sub][r];
#pragma unroll
      for (int off = 1; off < 16; off <<= 1)
#pragma unroll
        for (int r = 0; r < 8; ++r)
          rmax[r] = fmaxf(rmax[r], __shfl_xor(rmax[r], off, 16));

      float corr[8];
#pragma unroll
      for (int r = 0; r < 8; ++r) {
        const float mnew = fmaxf(mrow[sub][r], rmax[r]);
        corr[r]      = __expf(mrow[sub][r] - mnew);
        p[sub][r]    = __expf(p[sub][r] - mnew);
        mrow[sub][r] = mnew;
      }
      float rsum[8];
#pragma unroll
      for (int r = 0; r < 8; ++r) rsum[r] = p[sub][r];
#pragma unroll
      for (int off = 1; off < 16; off <<= 1)
#pragma unroll
        for (int r = 0; r < 8; ++r) rsum[r] += __shfl_xor(rsum[r], off, 16);
#pragma unroll
      for (int r = 0; r < 8; ++r)
        lrow[sub][r] = lrow[sub][r] * corr[r] + rsum[r];

      v8f* o = sub ? o1 : o0;
#pragma unroll
      for (int j = 0; j < 4; ++j)
#pragma unroll
        for (int r = 0; r < 8; ++r) o[j][r] *= corr[r];

      // C layout -> LDS (for A-layout reload)
#pragma unroll
      for (int r = 0; r < 8; ++r)
        pbuf[wave][sub][r + 8 * half][l15] = p[sub][r];
    }

    // ---- o += P @ V_tile : V fragments shared by both tiles ----
    const float* vbase = V + (size_t)s0 * K_HDIM;
#pragma unroll
    for (int kk = 0; kk < 4; ++kk) {
      const int kidx = 4 * kk + 2 * half;
      v2f pa0, pa1;
      pa0.x = pbuf[wave][0][l15][kidx];
      pa0.y = pbuf[wave][0][l15][kidx + 1];
      pa1.x = pbuf[wave][1][l15][kidx];
      pa1.y = pbuf[wave][1][l15][kidx + 1];
#pragma unroll
      for (int j = 0; j < 4; ++j) {
        v2f vf;
        vf.x = vbase[(size_t)kidx * K_HDIM + 16 * j + l15];
        vf.y = vbase[(size_t)(kidx + 1) * K_HDIM + 16 * j + l15];
        o0[j] = wmma_f32(pa0, vf, o0[j]);
        o1[j] = wmma_f32(pa1, vf, o1[j]);
      }
    }
  }

  // ---- normalize and store ctx as [B, T, D_MODEL] ----
#pragma unroll
  for (int sub = 0; sub < 2; ++sub) {
    const v8f* o = sub ? o1 : o0;
    float inv[8];
#pragma unroll
    for (int r = 0; r < 8; ++r) inv[r] = 1.0f / lrow[sub][r];
#pragma unroll
    for (int j = 0; j < 4; ++j)
#pragma unroll
      for (int r = 0; r < 8; ++r) {
        const int t   = t0 + sub * 16 + r + 8 * half;
        const int col = h * K_HDIM + 16 * j + l15;
        ctx[((size_t)b * K_T + t) * K_DMODEL + col] = o[j][r] * inv[r];
      }
  }
}

// =====================================================================
// Kernel 3: output projection  out = ctx @ Wo^T + bo  (32x32 per wave)
// =====================================================================
__global__ __launch_bounds__(128) void out_proj_kernel(
    const float* __restrict__ ctx, const float* __restrict__ Wo,
    const float* __restrict__ bo, float* __restrict__ out) {
  const int wave = threadIdx.x >> 5;
  const int lane = threadIdx.x & 31;
  const int half = lane >> 4;
  const int l15  = lane & 15;

  const int m0 = (blockIdx.x * 4 + wave) * 32;
  const int n0 = blockIdx.y * 32;

  const float* xr0 = ctx + (size_t)(m0 + l15) * K_DMODEL + 2 * half;
  const float* xr1 = xr0 + (size_t)16 * K_DMODEL;
  const float* wr0 = Wo + (size_t)(n0 + l15) * K_DMODEL + 2 * half;
  const float* wr1 = wr0 + (size_t)16 * K_DMODEL;

  v8f acc00 = {}, acc01 = {}, acc10 = {}, acc11 = {};
#pragma unroll 4
  for (int k0 = 0; k0 < K_DMODEL; k0 += 4) {
    v2f a0 = *(const v2f*)(xr0 + k0);
    v2f a1 = *(const v2f*)(xr1 + k0);
    v2f b0 = *(const v2f*)(wr0 + k0);
    v2f b1 = *(const v2f*)(wr1 + k0);
    acc00 = wmma_f32(a0, b0, acc00);
    acc01 = wmma_f32(a0, b1, acc01);
    acc10 = wmma_f32(a1, b0, acc10);
    acc11 = wmma_f32(a1, b1, acc11);
  }

#pragma unroll
  for (int mi = 0; mi < 2; ++mi) {
#pragma unroll
    for (int ni = 0; ni < 2; ++ni) {
      const v8f acc = (mi == 0) ? (ni == 0 ? acc00 : acc01)
                                : (ni == 0 ? acc10 : acc11);
      const int col = n0 + ni * 16 + l15;
      const float bn = bo[col];
#pragma unroll
      for (int r = 0; r < 8; ++r) {
        const int row = m0 + mi * 16 + r + 8 * half;
        out[(size_t)row * K_DMODEL + col] = acc[r] + bn;
      }
    }
  }
}

// =====================================================================
extern "C" void kernel_launch(void* const* d_in, const int* in_sizes, int n_in,
                              void* d_out, int out_size, void* d_ws,
                              size_t ws_size, hipStream_t stream) {
  const float* query = (const float*)d_in[0];
  const float* key_  = (const float*)d_in[1];
  const float* value = (const float*)d_in[2];
  const float* Wq    = (const float*)d_in[3];
  const float* bq    = (const float*)d_in[4];
  const float* Wk    = (const float*)d_in[5];
  const float* bk    = (const float*)d_in[6];
  const float* Wv    = (const float*)d_in[7];
  const float* bv    = (const float*)d_in[8];
  const float* Wo    = (const float*)d_in[9];
  const float* bo    = (const float*)d_in[10];
  const float* rel   = (const float*)d_in[11];
  float* out = (float*)d_out;

  const size_t per = (size_t)K_B * K_NHEAD * K_T * K_HDIM;  // 4,194,304 floats
  float* qb  = (float*)d_ws;
  float* kb  = qb + per;
  float* vb  = kb + per;
  float* ctx = vb + per;  // total 64 MB of workspace

  // QKV projections: 8192/32 row-tiles / 4 waves, 512/32 col-tiles, 3 mats
  dim3 gp((K_B * K_T) / (32 * 4), K_DMODEL / 32, 3);
  proj_qkv_kernel<<<gp, 128, 0, stream>>>(query, key_, value, Wq, bq, Wk, bk,
                                          Wv, bv, qb, kb, vb);

  // Flash attention: 32 (b,h) x 64 row-pair tiles = 2048 waves / 4 per block
  dim3 ga((K_B * K_NHEAD * (K_T / 32)) / 4);
  flash_attn_kernel<<<ga, 128, 0, stream>>>(qb, kb, vb, rel, ctx);

  // Output projection
  dim3 go((K_B * K_T) / (32 * 4), K_DMODEL / 32);
  out_proj_kernel<<<go, 128, 0, stream>>>(ctx, Wo, bo, out);
}